// SAELogicAgentV2_63350767616572
// MI455X (gfx1250) — compile-verified
//
#include <hip/hip_runtime.h>
#include <math.h>

#define B_ROWS   1024
#define D_IN     1024
#define D_HID    2048
#define K_TOP    64
#define N_CLAUSE 64

typedef __attribute__((ext_vector_type(2))) float v2f;
typedef __attribute__((ext_vector_type(8))) float v8f;

__device__ __forceinline__ float softplusf_(float x) {
    return fmaxf(x, 0.f) + log1pf(expf(-fabsf(x)));
}
__device__ __forceinline__ float sigmoidf_(float x) {
    return 1.f / (1.f + expf(-x));
}

// ---------------------------------------------------------------------------
// Kernel 1: per-(clause,feature) table, packed float4 {A, D, L0, 0} at
// P[h*64 + c]  (feature-major -> clause gathers are a single contiguous b128).
//   A = 1 - p + 1e-8,  D = p - n,  L0 = log(A + D*f0),  f0 = sigmoid(-alpha*beta)
// ---------------------------------------------------------------------------
__global__ __launch_bounds__(256) void prep_tables(
        const float* __restrict__ w_pos, const float* __restrict__ w_neg,
        const float* __restrict__ log_alpha, const float* __restrict__ beta,
        float4* __restrict__ P) {
    int tid = blockIdx.x * blockDim.x + threadIdx.x;   // 0 .. 64*2048-1
    int c = tid >> 11;            // clause
    int h = tid & (D_HID - 1);    // feature (fastest -> coalesced reads)
    float wp = w_pos[c * D_HID + h];
    float wn = w_neg[c * D_HID + h];
    float m  = fmaxf(fmaxf(wp, wn), 0.f);
    float ep = expf(wp - m), en = expf(wn - m), ez = expf(0.f - m);
    float inv = 1.f / (ep + en + ez);
    float p = ep * inv, n = en * inv;
    float A = 1.f - p + 1e-8f;
    float D = p - n;
    float alpha = softplusf_(log_alpha[h]) + 0.5f;
    float f0 = sigmoidf_(-alpha * beta[h]);
    float L0 = logf(A + D * f0);
    P[h * N_CLAUSE + c] = make_float4(A, D, L0, 0.f);
}

// ---------------------------------------------------------------------------
// Kernel 2: base[c] = sum_h L0[c,h]
// ---------------------------------------------------------------------------
__global__ __launch_bounds__(256) void base_reduce(
        const float4* __restrict__ P, float* __restrict__ base) {
    int c = blockIdx.x;
    __shared__ float red[256];
    float s = 0.f;
    for (int h = threadIdx.x; h < D_HID; h += 256) s += P[h * N_CLAUSE + c].z;
    red[threadIdx.x] = s; __syncthreads();
    for (int off = 128; off > 0; off >>= 1) {
        if (threadIdx.x < off) red[threadIdx.x] += red[threadIdx.x + off];
        __syncthreads();
    }
    if (threadIdx.x == 0) base[c] = red[0];
}

// ---------------------------------------------------------------------------
// Kernel 3: z = relu(x @ W + b), fp32 WMMA 16x16x4.
// 128 threads (4 waves), block tile 64x64, wave tile 32x32 (2x2 accumulators),
// K-chunk 32, register-staged double-buffered LDS.
// ---------------------------------------------------------------------------
#define TM 64
#define TN 64
#define TK 32

__global__ __launch_bounds__(128) void gemm_relu(
        const float* __restrict__ X, const float* __restrict__ W,
        const float* __restrict__ bias, float* __restrict__ Z) {
    __shared__ float As[2][TK][TM + 1];   // [buf][k][m]
    __shared__ float Bs[2][TK][TN + 1];   // [buf][k][n]
    const int t    = threadIdx.x;
    const int lane = t & 31;
    const int wv   = t >> 5;
    const int half = lane >> 4;           // 0: lanes 0-15, 1: lanes 16-31
    const int ln   = lane & 15;
    const int wr   = (wv >> 1) * 32;      // wave M offset in block tile
    const int wc   = (wv & 1) * 32;       // wave N offset in block tile
    const int mBase = blockIdx.y * TM;
    const int nBase = blockIdx.x * TN;

    v8f acc[2][2] = {};
    float areg[16], breg[16];

    // prologue: load chunk 0 -> regs -> LDS buf 0
    #pragma unroll
    for (int i = 0; i < 16; i++) {
        int idx = t + i * 128;
        areg[i] = X[(mBase + (idx >> 5)) * D_IN + (idx & 31)];          // coalesced over k
        breg[i] = W[((idx >> 6)) * D_HID + nBase + (idx & 63)];         // coalesced over n
    }
    #pragma unroll
    for (int i = 0; i < 16; i++) {
        int idx = t + i * 128;
        As[0][idx & 31][idx >> 5] = areg[i];
        Bs[0][idx >> 6][idx & 63] = breg[i];
    }
    __syncthreads();

    int buf = 0;
    for (int k0 = TK; k0 <= D_IN; k0 += TK) {
        const bool last = (k0 == D_IN);
        if (!last) {   // prefetch next chunk into registers (overlaps compute)
            #pragma unroll
            for (int i = 0; i < 16; i++) {
                int idx = t + i * 128;
                areg[i] = X[(mBase + (idx >> 5)) * D_IN + k0 + (idx & 31)];
                breg[i] = W[(k0 + (idx >> 6)) * D_HID + nBase + (idx & 63)];
            }
        }
        #pragma unroll
        for (int kk = 0; kk < TK; kk += 4) {
            const int kA = kk + 2 * half;   // lanes<16: K{0,1}; lanes>=16: K{2,3}
            v2f a0, a1, b0, b1;
            a0.x = As[buf][kA][wr + ln];          a0.y = As[buf][kA + 1][wr + ln];
            a1.x = As[buf][kA][wr + 16 + ln];     a1.y = As[buf][kA + 1][wr + 16 + ln];
            b0.x = Bs[buf][kA][wc + ln];          b0.y = Bs[buf][kA + 1][wc + ln];
            b1.x = Bs[buf][kA][wc + 16 + ln];     b1.y = Bs[buf][kA + 1][wc + 16 + ln];
            acc[0][0] = __builtin_amdgcn_wmma_f32_16x16x4_f32(false, a0, false, b0, (short)0, acc[0][0], false, false);
            acc[0][1] = __builtin_amdgcn_wmma_f32_16x16x4_f32(false, a0, false, b1, (short)0, acc[0][1], false, false);
            acc[1][0] = __builtin_amdgcn_wmma_f32_16x16x4_f32(false, a1, false, b0, (short)0, acc[1][0], false, false);
            acc[1][1] = __builtin_amdgcn_wmma_f32_16x16x4_f32(false, a1, false, b1, (short)0, acc[1][1], false, false);
        }
        if (!last) {
            __syncthreads();
            buf ^= 1;
            #pragma unroll
            for (int i = 0; i < 16; i++) {
                int idx = t + i * 128;
                As[buf][idx & 31][idx >> 5] = areg[i];
                Bs[buf][idx >> 6][idx & 63] = breg[i];
            }
            __syncthreads();
        }
    }

    #pragma unroll
    for (int i = 0; i < 2; i++) {
        #pragma unroll
        for (int j = 0; j < 2; j++) {
            const int col = nBase + wc + j * 16 + ln;
            const float bv = bias[col];
            #pragma unroll
            for (int v = 0; v < 8; v++) {
                const int row = mBase + wr + i * 16 + half * 8 + v;  // C VGPR v: M=v (+8 hi half)
                float val = acc[i][j][v] + bv;
                Z[row * D_HID + col] = (val > 0.f) ? val : 0.f;       // force +0 bits
            }
        }
    }
}

// ---------------------------------------------------------------------------
// Kernel 4: exact top-64 per row: bitwise binary search on IEEE bits
// (monotone for >=0 floats; bit31 skipped since all values are +0 or positive),
// deterministic prefix-scan compaction.
// ---------------------------------------------------------------------------
__global__ __launch_bounds__(256) void topk_select(
        const float* __restrict__ Z, int* __restrict__ tidx, float* __restrict__ tval) {
    const int row = blockIdx.x;
    const int t = threadIdx.x;
    const float4* zrow = (const float4*)(Z + (size_t)row * D_HID);
    float4 za = zrow[t];          // elements 4t .. 4t+3
    float4 zb = zrow[t + 256];    // elements 1024+4t .. 1024+4t+3
    unsigned v[8];
    v[0] = __float_as_uint(za.x); v[1] = __float_as_uint(za.y);
    v[2] = __float_as_uint(za.z); v[3] = __float_as_uint(za.w);
    v[4] = __float_as_uint(zb.x); v[5] = __float_as_uint(zb.y);
    v[6] = __float_as_uint(zb.z); v[7] = __float_as_uint(zb.w);

    __shared__ int red[8];
    __shared__ int sh_total;
    __shared__ int sgt[256], seq[256];

    unsigned thresh = 0u;
    for (int bit = 30; bit >= 0; --bit) {   // bit31 (sign) is always 0
        unsigned cand = thresh | (1u << bit);
        int cnt = 0;
        #pragma unroll
        for (int j = 0; j < 8; j++) cnt += (v[j] >= cand) ? 1 : 0;
        for (int off = 16; off > 0; off >>= 1) cnt += __shfl_xor(cnt, off, 32);
        if ((t & 31) == 0) red[t >> 5] = cnt;
        __syncthreads();
        if (t == 0) {
            int tot = 0;
            for (int i = 0; i < 8; i++) tot += red[i];
            sh_total = tot;
        }
        __syncthreads();
        if (sh_total >= K_TOP) thresh = cand;
        __syncthreads();
    }
    // thresh == exact K-th largest value's bit pattern
    int cgt = 0, ceq = 0;
    #pragma unroll
    for (int j = 0; j < 8; j++) { cgt += (v[j] > thresh); ceq += (v[j] == thresh); }
    sgt[t] = cgt; seq[t] = ceq; __syncthreads();
    for (int off = 1; off < 256; off <<= 1) {
        int a = sgt[t], b = seq[t];
        int a2 = (t >= off) ? sgt[t - off] : 0;
        int b2 = (t >= off) ? seq[t - off] : 0;
        __syncthreads();
        sgt[t] = a + a2; seq[t] = b + b2;
        __syncthreads();
    }
    int gtExcl  = sgt[t] - cgt;
    int eqExcl  = seq[t] - ceq;
    int totalGt = sgt[255];
    int lg = 0, le = 0;
    #pragma unroll
    for (int j = 0; j < 8; j++) {
        int h = (j < 4) ? (4 * t + j) : (1024 + 4 * t + (j - 4));
        if (v[j] > thresh) {
            int slot = gtExcl + (lg++);
            if (slot < K_TOP) {
                tidx[row * K_TOP + slot] = h;
                tval[row * K_TOP + slot] = __uint_as_float(v[j]);
            }
        } else if (v[j] == thresh) {
            int slot = totalGt + eqExcl + (le++);
            if (slot < K_TOP) {
                tidx[row * K_TOP + slot] = h;
                tval[row * K_TOP + slot] = __uint_as_float(v[j]);
            }
        }
    }
}

// ---------------------------------------------------------------------------
// Kernel 5: clause corrections + sigmoid + action reduce.  64 threads/row.
// Per (active feature, clause-thread): one b128 gather + one log.
// ---------------------------------------------------------------------------
__global__ __launch_bounds__(64) void clause_kernel(
        const int* __restrict__ tidx, const float* __restrict__ tval,
        const float* __restrict__ log_alpha, const float* __restrict__ beta,
        const float4* __restrict__ P, const float* __restrict__ base,
        const float* __restrict__ cw, float* __restrict__ out) {
    const int row = blockIdx.x;
    const int c = threadIdx.x;
    __shared__ int   sh_h[K_TOP];
    __shared__ float sh_f[K_TOP];
    __shared__ float sh_cl[N_CLAUSE];
    {
        int   h = tidx[row * K_TOP + c];
        float v = tval[row * K_TOP + c];
        float alpha = softplusf_(log_alpha[h]) + 0.5f;
        sh_h[c] = h;
        sh_f[c] = sigmoidf_(alpha * (v - beta[h]));
    }
    __syncthreads();
    float acc = 0.f;
    for (int k = 0; k < K_TOP; k++) {        // fixed order -> deterministic
        int   h = sh_h[k];
        float f = sh_f[k];
        float4 q = P[h * N_CLAUSE + c];      // contiguous 1KB across 64 threads
        acc += logf(q.x + q.y * f) - q.z;
    }
    sh_cl[c] = sigmoidf_(acc + base[c] + cw[c]);
    __syncthreads();
    if (c < 8) {
        float s = 0.f;
        #pragma unroll
        for (int j = 0; j < 8; j++) s += sh_cl[c * 8 + j];
        out[row * 8 + c] = s;
    }
}

// ---------------------------------------------------------------------------
extern "C" void kernel_launch(void* const* d_in, const int* in_sizes, int n_in,
                              void* d_out, int out_size, void* d_ws, size_t ws_size,
                              hipStream_t stream) {
    (void)in_sizes; (void)n_in; (void)out_size; (void)ws_size;
    const float* x         = (const float*)d_in[0];
    const float* W_enc     = (const float*)d_in[1];
    const float* b_enc     = (const float*)d_in[2];
    const float* log_alpha = (const float*)d_in[3];
    const float* beta      = (const float*)d_in[4];
    const float* w_pos     = (const float*)d_in[5];
    const float* w_neg     = (const float*)d_in[6];
    const float* cw        = (const float*)d_in[7];
    float* out = (float*)d_out;

    char* ws = (char*)d_ws;
    float*  z    = (float*)ws;  ws += (size_t)B_ROWS * D_HID * sizeof(float);      // 8 MB
    float4* P    = (float4*)ws; ws += (size_t)D_HID * N_CLAUSE * sizeof(float4);   // 2 MB
    float*  base = (float*)ws;  ws += (size_t)N_CLAUSE * sizeof(float);
    int*    tix  = (int*)ws;    ws += (size_t)B_ROWS * K_TOP * sizeof(int);
    float*  tv   = (float*)ws;  ws += (size_t)B_ROWS * K_TOP * sizeof(float);

    prep_tables<<<(N_CLAUSE * D_HID) / 256, 256, 0, stream>>>(
        w_pos, w_neg, log_alpha, beta, P);
    base_reduce<<<N_CLAUSE, 256, 0, stream>>>(P, base);
    gemm_relu<<<dim3(D_HID / TN, B_ROWS / TM), 128, 0, stream>>>(x, W_enc, b_enc, z);
    topk_select<<<B_ROWS, 256, 0, stream>>>(z, tix, tv);
    clause_kernel<<<B_ROWS, N_CLAUSE, 0, stream>>>(
        tix, tv, log_alpha, beta, P, base, cw, out);
}